// WeightedKLSimilarityLoss_70772471103984
// MI455X (gfx1250) — compile-verified
//
#include <hip/hip_runtime.h>
#include <math.h>

// ---------------------------------------------------------------------------
// WeightedKLSimilarityLoss, fused for MI455X (gfx1250, wave32, WMMA).
//
// Never materialize the 8192x8192 sim matrices (2 x 268MB, ~1.6GB HBM).
// Inputs (42MB) stay L2-resident; sim tiles are recomputed in two sweeps
// (row-sum stats, then loss) with V_WMMA_F32_16X16X4_F32.
// 2x2 register tiling per wave: 4 fragment loads -> 4 WMMAs per k-step
// (1.0 loads/WMMA instead of 2.0), 4 independent accumulator chains.
// Cos-sims bounded in [-1,1] => fixed softmax shift of 1.0 (exactly
// equivalent to the reference's row-max by shift invariance).
// ---------------------------------------------------------------------------

typedef float v2f __attribute__((ext_vector_type(2)));
typedef float v8f __attribute__((ext_vector_type(8)));

#define N_ROWS   8192
#define D_O      512
#define D_T      768
#define NSTRIP   16                    // column strips (grid.y)
#define STRIP_W  (N_ROWS / NSTRIP)     // 512 columns per strip
#define CTILES   (STRIP_W / 32)        // 16 col tiles (32 wide) per strip
#define WAVES    4                     // 128-thread block = 4 wave32

__device__ __forceinline__ v8f wmma4(v2f a, v2f b, v8f c) {
    return __builtin_amdgcn_wmma_f32_16x16x4_f32(
        /*neg_a=*/false, a, /*neg_b=*/false, b,
        /*c_mod=*/(short)0, c, /*reuse_a=*/false, /*reuse_b=*/false);
}

// 32x32 raw Gram tile block (2x2 of 16x16):  C[ri][cj] = X[i0+16ri,:] X[j0+16cj,:]^T
// Fragment layout (f32 16x16x4): lane m=lane&15, h=lane>>4 holds
//   {X[r0+m][k+2h], X[r0+m][k+2h+1]}  for both A and B (B = other-rows^T).
__device__ __forceinline__ void gram32(const float* __restrict__ X, int D,
                                       int i0, int j0, int lane, v8f c[2][2]) {
    const int m = lane & 15;
    const int h = lane >> 4;
    const float* pa0 = X + (size_t)(i0 + m) * D + 2 * h;
    const float* pa1 = pa0 + (size_t)16 * D;
    const float* pb0 = X + (size_t)(j0 + m) * D + 2 * h;
    const float* pb1 = pb0 + (size_t)16 * D;
#pragma unroll 4
    for (int k = 0; k < D; k += 4) {
        v2f a0 = *(const v2f*)(pa0 + k);
        v2f a1 = *(const v2f*)(pa1 + k);
        v2f b0 = *(const v2f*)(pb0 + k);
        v2f b1 = *(const v2f*)(pb1 + k);
        c[0][0] = wmma4(a0, b0, c[0][0]);
        c[0][1] = wmma4(a0, b1, c[0][1]);
        c[1][0] = wmma4(a1, b0, c[1][0]);
        c[1][1] = wmma4(a1, b1, c[1][1]);
    }
}

// ---------------------------------------------------------------------------
__global__ void init_kernel(float* Zo, float* Zt, float* acc) {
    int i = blockIdx.x * blockDim.x + threadIdx.x;
    if (i < N_ROWS) { Zo[i] = 0.0f; Zt[i] = 0.0f; }
    if (i == 0) acc[0] = 0.0f;
}

// One wave per row: inv_norm[i] = 1 / max(||X_i||, 1e-8)
__global__ void norm_kernel(const float* __restrict__ X, int D,
                            float* __restrict__ rn) {
    const int row  = blockIdx.x * (blockDim.x >> 5) + (threadIdx.x >> 5);
    const int lane = threadIdx.x & 31;
    if (row >= N_ROWS) return;
    const float* p = X + (size_t)row * D;
    float s = 0.0f;
    for (int k = lane; k < D; k += 32) { float v = p[k]; s += v * v; }
#pragma unroll
    for (int m = 1; m < 32; m <<= 1) s += __shfl_xor(s, m, 32);
    if (lane == 0) rn[row] = 1.0f / fmaxf(sqrtf(s), 1e-8f);
}

// Pass 1: Zo[i] = sum_{j!=i} exp(cos_o(i,j) - 1), Zt likewise.
__global__ __launch_bounds__(128)
void stats_kernel(const float* __restrict__ Xo, const float* __restrict__ Xt,
                  const float* __restrict__ rno, const float* __restrict__ rnt,
                  float* __restrict__ Zo, float* __restrict__ Zt) {
    const int i0    = blockIdx.x * 32;
    const int strip = blockIdx.y;
    const int w     = threadIdx.x >> 5;
    const int lane  = threadIdx.x & 31;
    const int h     = lane >> 4;
    const int nlo   = lane & 15;

    float rio[2][8], rit[2][8];
#pragma unroll
    for (int ri = 0; ri < 2; ++ri) {
        const int rb = i0 + 16 * ri + 8 * h;
#pragma unroll
        for (int r = 0; r < 8; ++r) { rio[ri][r] = rno[rb + r]; rit[ri][r] = rnt[rb + r]; }
    }

    float accO[2][8] = {}, accT[2][8] = {};

    for (int jt = w; jt < CTILES; jt += WAVES) {
        const int j0 = strip * STRIP_W + jt * 32;
        v8f co[2][2] = {}; gram32(Xo, D_O, i0, j0, lane, co);
        v8f ct[2][2] = {}; gram32(Xt, D_T, i0, j0, lane, ct);
#pragma unroll
        for (int cj = 0; cj < 2; ++cj) {
            const int   col = j0 + 16 * cj + nlo;
            const float rjo = rno[col];
            const float rjt = rnt[col];
#pragma unroll
            for (int ri = 0; ri < 2; ++ri) {
                const int rb = i0 + 16 * ri + 8 * h;
#pragma unroll
                for (int r = 0; r < 8; ++r) {
                    if (rb + r != col) {              // diagonal: teacher prob 0
                        accO[ri][r] += __expf(co[ri][cj][r] * rio[ri][r] * rjo - 1.0f);
                        accT[ri][r] += __expf(ct[ri][cj][r] * rit[ri][r] * rjt - 1.0f);
                    }
                }
            }
        }
    }
#pragma unroll
    for (int ri = 0; ri < 2; ++ri) {
        const int rb = i0 + 16 * ri + 8 * h;
#pragma unroll
        for (int r = 0; r < 8; ++r) {
            float vo = accO[ri][r], vt = accT[ri][r];
#pragma unroll
            for (int m = 1; m < 16; m <<= 1) {        // reduce 16 lanes per half
                vo += __shfl_xor(vo, m, 32);
                vt += __shfl_xor(vt, m, 32);
            }
            if (nlo == 0) {
                atomicAdd(&Zo[rb + r], vo);
                atomicAdd(&Zt[rb + r], vt);
            }
        }
    }
}

// Pass 2: sum_{i!=j} t^3 * (log t - log s),  t = exp(st-1)/Zt,
// log t = (st-1)-log Zt, log s = (so-1)-log Zo.
__global__ __launch_bounds__(128)
void loss_kernel(const float* __restrict__ Xo, const float* __restrict__ Xt,
                 const float* __restrict__ rno, const float* __restrict__ rnt,
                 const float* __restrict__ Zo, const float* __restrict__ Zt,
                 float* __restrict__ acc) {
    const int i0    = blockIdx.x * 32;
    const int strip = blockIdx.y;
    const int w     = threadIdx.x >> 5;
    const int lane  = threadIdx.x & 31;
    const int h     = lane >> 4;
    const int nlo   = lane & 15;

    float rio[2][8], rit[2][8], lzo[2][8], lzt[2][8], izt[2][8];
#pragma unroll
    for (int ri = 0; ri < 2; ++ri) {
        const int rb = i0 + 16 * ri + 8 * h;
#pragma unroll
        for (int r = 0; r < 8; ++r) {
            rio[ri][r] = rno[rb + r];
            rit[ri][r] = rnt[rb + r];
            float zo = Zo[rb + r], zt = Zt[rb + r];
            lzo[ri][r] = __logf(zo);
            lzt[ri][r] = __logf(zt);
            izt[ri][r] = 1.0f / zt;
        }
    }

    float lsum = 0.0f;
    for (int jt = w; jt < CTILES; jt += WAVES) {
        const int j0 = strip * STRIP_W + jt * 32;
        v8f co[2][2] = {}; gram32(Xo, D_O, i0, j0, lane, co);
        v8f ct[2][2] = {}; gram32(Xt, D_T, i0, j0, lane, ct);
#pragma unroll
        for (int cj = 0; cj < 2; ++cj) {
            const int   col = j0 + 16 * cj + nlo;
            const float rjo = rno[col];
            const float rjt = rnt[col];
#pragma unroll
            for (int ri = 0; ri < 2; ++ri) {
                const int rb = i0 + 16 * ri + 8 * h;
#pragma unroll
                for (int r = 0; r < 8; ++r) {
                    if (rb + r != col) {
                        float so = co[ri][cj][r] * rio[ri][r] * rjo - 1.0f; // shifted student logit
                        float st = ct[ri][cj][r] * rit[ri][r] * rjt - 1.0f; // shifted teacher logit
                        float t    = __expf(st) * izt[ri][r];               // teacher prob
                        float tlog = st - lzt[ri][r];                       // log teacher
                        float slog = so - lzo[ri][r];                       // student log-prob
                        lsum += t * t * t * (tlog - slog);
                    }
                }
            }
        }
    }
#pragma unroll
    for (int m = 1; m < 32; m <<= 1) lsum += __shfl_xor(lsum, m, 32);
    if (lane == 0) atomicAdd(acc, lsum);
}

__global__ void finalize_kernel(const float* __restrict__ acc,
                                float* __restrict__ out) {
    if (threadIdx.x == 0)
        out[0] = acc[0] * (1.0f / ((float)N_ROWS * (float)N_ROWS));  // WEIGHT=1
}

// ---------------------------------------------------------------------------
extern "C" void kernel_launch(void* const* d_in, const int* in_sizes, int n_in,
                              void* d_out, int out_size, void* d_ws, size_t ws_size,
                              hipStream_t stream) {
    const float* Xo = (const float*)d_in[0];   // [8192, 512]
    const float* Xt = (const float*)d_in[1];   // [8192, 768]
    float* ws  = (float*)d_ws;
    float* rno = ws;                 // 8192
    float* rnt = ws + N_ROWS;        // 8192
    float* Zo  = ws + 2 * N_ROWS;    // 8192
    float* Zt  = ws + 3 * N_ROWS;    // 8192
    float* acc = ws + 4 * N_ROWS;    // 1

    init_kernel<<<(N_ROWS + 255) / 256, 256, 0, stream>>>(Zo, Zt, acc);
    norm_kernel<<<N_ROWS / 8, 256, 0, stream>>>(Xo, D_O, rno);
    norm_kernel<<<N_ROWS / 8, 256, 0, stream>>>(Xt, D_T, rnt);

    dim3 grid(N_ROWS / 32, NSTRIP);  // 256 row blocks x 16 column strips
    stats_kernel<<<grid, 128, 0, stream>>>(Xo, Xt, rno, rnt, Zo, Zt);
    loss_kernel <<<grid, 128, 0, stream>>>(Xo, Xt, rno, rnt, Zo, Zt, acc);

    finalize_kernel<<<1, 32, 0, stream>>>(acc, (float*)d_out);
}